// TeamPerformanceGNN_13340168422064
// MI455X (gfx1250) — compile-verified
//
#include <hip/hip_runtime.h>
#include <hip/hip_bf16.h>
#include <stdint.h>

#define FN_ 128
#define FE_ 16
#define H_  64
#define G_  64
#define NEG_SLOPE 0.2f

typedef __attribute__((ext_vector_type(2)))  float    v2f;
typedef __attribute__((ext_vector_type(8)))  float    v8f;

// ---------------- utility: zero fill ----------------
__global__ void k_zero(float* __restrict__ p, size_t n) {
  size_t i = (size_t)blockIdx.x * blockDim.x + threadIdx.x;
  size_t s = (size_t)gridDim.x * blockDim.x;
  for (; i < n; i += s) p[i] = 0.f;
}

// ---------------- edge_attr column mean ----------------
__global__ void k_mean_accum(const float* __restrict__ ea, int E, float* __restrict__ mean) {
  __shared__ float sm[FE_];
  const int t = threadIdx.x;
  if (t < FE_) sm[t] = 0.f;
  __syncthreads();
  float loc[FE_];
#pragma unroll
  for (int k = 0; k < FE_; ++k) loc[k] = 0.f;
  for (int row = blockIdx.x * blockDim.x + t; row < E; row += gridDim.x * blockDim.x) {
    const float* r = ea + (size_t)row * FE_;
#pragma unroll
    for (int k = 0; k < FE_; ++k) loc[k] += r[k];
  }
#pragma unroll
  for (int k = 0; k < FE_; ++k) atomicAdd(&sm[k], loc[k]);   // ds_add_f32
  __syncthreads();
  if (t < FE_) atomicAdd(&mean[t], sm[t]);
}

__global__ void k_mean_scale(float* mean, int E) {
  int t = threadIdx.x;
  if (t < FE_) mean[t] /= (float)E;
}

// ---------------- fused dual WMMA GEMM ----------------
// Cl[M,64] = A[M,K] * Bl[K,64];  Cr[M,64] = A[M,K] * Br[K,64]   (row major, fp32)
// One wave per 16-row block; covers all 4 column tiles for BOTH outputs:
// A fragment loaded once per k-step, feeds 8 v_wmma_f32_16x16x4_f32.
// Requires gridDim.x == M/16 full tiles (tail rows handled by k_gemm2_tail).
template <int K>
__global__ __launch_bounds__(32)
void k_gemm2_wmma(const float* __restrict__ A,
                  const float* __restrict__ Bl, const float* __restrict__ Br,
                  float* __restrict__ Cl, float* __restrict__ Cr) {
  const int row0 = blockIdx.x << 4;
  const int lane = threadIdx.x;
  const int half = lane >> 4;   // lanes 16-31 carry K pair {2,3} per fp32 A/B layout
  const int idx  = lane & 15;   // A: M row; B/C/D: N column
  const float* __restrict__ Arow = A + (size_t)(row0 + idx) * K;
  const float* __restrict__ BlBase = Bl + idx;
  const float* __restrict__ BrBase = Br + idx;
  __builtin_prefetch(Arow, 0, 3);            // global_prefetch_b8
  v8f accl[4] = {}, accr[4] = {};
#pragma unroll
  for (int k = 0; k < K; k += 4) {
    const int ka = k + (half << 1);
    v2f a;
    a.x = Arow[ka];
    a.y = Arow[ka + 1];
#pragma unroll
    for (int t = 0; t < 4; ++t) {
      const int c = t << 4;
      v2f bl, br;
      bl.x = BlBase[ka * H_ + c];
      bl.y = BlBase[(ka + 1) * H_ + c];
      br.x = BrBase[ka * H_ + c];
      br.y = BrBase[(ka + 1) * H_ + c];
      accl[t] = __builtin_amdgcn_wmma_f32_16x16x4_f32(false, a, false, bl, (short)0, accl[t], false, false);
      accr[t] = __builtin_amdgcn_wmma_f32_16x16x4_f32(false, a, false, br, (short)0, accr[t], false, false);
    }
  }
  float* cl = Cl + (size_t)(row0 + (half << 3)) * H_ + idx;
  float* cr = Cr + (size_t)(row0 + (half << 3)) * H_ + idx;
#pragma unroll
  for (int t = 0; t < 4; ++t) {
#pragma unroll
    for (int r = 0; r < 8; ++r) {
      cl[r * H_ + (t << 4)] = accl[t][r];
      cr[r * H_ + (t << 4)] = accr[t][r];
    }
  }
}

// scalar cleanup for M % 16 tail rows (not hit for N=50000, kept for safety)
__global__ void k_gemm2_tail(const float* __restrict__ A,
                             const float* __restrict__ Bl, const float* __restrict__ Br,
                             float* __restrict__ Cl, float* __restrict__ Cr,
                             int M, int K, int row_start) {
  const int i = row_start + blockIdx.x;
  const int c = threadIdx.x;
  if (i >= M || c >= H_) return;
  float sl = 0.f, sr = 0.f;
  for (int k = 0; k < K; ++k) {
    const float av = A[(size_t)i * K + k];
    sl = fmaf(av, Bl[k * H_ + c], sl);
    sr = fmaf(av, Br[k * H_ + c], sr);
  }
  Cl[(size_t)i * H_ + c] = sl;
  Cr[(size_t)i * H_ + c] = sr;
}

// ---------------- ordered-uint float max encoding ----------------
__device__ __forceinline__ unsigned enc_f32(float f) {
  unsigned u = __float_as_uint(f);
  return (u & 0x80000000u) ? ~u : (u | 0x80000000u);
}
__device__ __forceinline__ float dec_f32(unsigned u) {
  return (u & 0x80000000u) ? __uint_as_float(u & 0x7fffffffu) : __uint_as_float(~u);
}

// ---------------- edge score: s = att . LeakyReLU(xl[src]+xr[dst]+ea@We); segment max ----------------
__global__ void k_edge_score(const float* __restrict__ xl, const float* __restrict__ xr,
                             const float* __restrict__ ea, const float* __restrict__ mean,
                             const float* __restrict__ We, const float* __restrict__ att,
                             const int* __restrict__ ei, int E, int EP,
                             float* __restrict__ sc, unsigned* __restrict__ nmax) {
  const int e = blockIdx.x * blockDim.x + threadIdx.x;
  if (e >= EP) return;
  const int  self = e - E;
  const int  src  = (e < E) ? ei[e]     : self;
  const int  dst  = (e < E) ? ei[E + e] : self;
  const float* earow = (e < E) ? (ea + (size_t)e * FE_) : mean;
  float eav[FE_];
#pragma unroll
  for (int k = 0; k < FE_; ++k) eav[k] = earow[k];
  const float* xls = xl + (size_t)src * H_;
  const float* xrd = xr + (size_t)dst * H_;
  __builtin_prefetch(xls, 0, 3);
  __builtin_prefetch(xrd, 0, 3);
  float s = 0.f;
#pragma unroll 4
  for (int h = 0; h < H_; ++h) {
    float m = xls[h] + xrd[h];
#pragma unroll
    for (int k = 0; k < FE_; ++k) m = fmaf(eav[k], We[k * H_ + h], m);
    m = (m > 0.f) ? m : NEG_SLOPE * m;       // LeakyReLU
    s = fmaf(m, att[h], s);
  }
  sc[e] = s;
  atomicMax(&nmax[dst], enc_f32(s));
}

// ---------------- softmax numerator + segment sum ----------------
__global__ void k_edge_exp(float* __restrict__ sc, const unsigned* __restrict__ nmax,
                           float* __restrict__ nsum, const int* __restrict__ ei,
                           int E, int EP) {
  const int e = blockIdx.x * blockDim.x + threadIdx.x;
  if (e >= EP) return;
  const int dst = (e < E) ? ei[E + e] : (e - E);
  const float ex = __expf(sc[e] - dec_f32(nmax[dst]));
  sc[e] = ex;
  atomicAdd(&nsum[dst], ex);
}

// ---------------- aggregation: acc[dst] += alpha * xl[src]; one 32-lane wave per edge ----------------
__global__ void k_edge_agg(const float* __restrict__ xl, const float* __restrict__ sc,
                           const float* __restrict__ nsum, const int* __restrict__ ei,
                           int E, int EP, float* __restrict__ acc) {
  const int e    = blockIdx.x * 8 + (threadIdx.x >> 5);
  const int lane = threadIdx.x & 31;
  if (e >= EP) return;
  const int self = e - E;
  const int src  = (e < E) ? ei[e]     : self;
  const int dst  = (e < E) ? ei[E + e] : self;
  const float alpha = sc[e] / nsum[dst];
  const float* xls = xl + (size_t)src * H_;
  float* ad = acc + (size_t)dst * H_;
  atomicAdd(&ad[lane],      alpha * xls[lane]);
  atomicAdd(&ad[lane + 32], alpha * xls[lane + 32]);
}

// ---------------- bias + ReLU (in place) ----------------
__global__ void k_finalize(float* __restrict__ p, const float* __restrict__ b, size_t n) {
  const size_t i = (size_t)blockIdx.x * blockDim.x + threadIdx.x;
  if (i >= n) return;
  const float v = p[i] + b[i & (H_ - 1)];
  p[i] = v > 0.f ? v : 0.f;
}

// ---------------- node prediction head ----------------
__global__ void k_node_pred(const float* __restrict__ h, const float* __restrict__ Wn,
                            const float* __restrict__ bn, float* __restrict__ out, int N) {
  const int i = blockIdx.x * blockDim.x + threadIdx.x;
  if (i >= N) return;
  const float* hi = h + (size_t)i * H_;
  float s = 0.f;
#pragma unroll
  for (int k = 0; k < H_; ++k) s = fmaf(hi[k], Wn[k], s);
  out[i] = s + bn[0];
}

// ---------------- global mean pool: accumulate ----------------
__global__ void k_pool(const float* __restrict__ h, const int* __restrict__ batch,
                       float* __restrict__ gsum, float* __restrict__ gcnt, int N) {
  const int i    = blockIdx.x * 8 + (threadIdx.x >> 5);
  const int lane = threadIdx.x & 31;
  if (i >= N) return;
  const int g = batch[i];
  const float* hi = h + (size_t)i * H_;
  float* gs = gsum + (size_t)g * H_;
  atomicAdd(&gs[lane],      hi[lane]);
  atomicAdd(&gs[lane + 32], hi[lane + 32]);
  if (lane == 0) atomicAdd(&gcnt[g], 1.f);
}

// ---------------- graph prediction head ----------------
__global__ void k_graph_pred(const float* __restrict__ gsum, const float* __restrict__ gcnt,
                             const float* __restrict__ Wg, const float* __restrict__ bg,
                             float* __restrict__ out, int N, int G) {
  const int g = threadIdx.x;
  if (g >= G) return;
  const float c = fmaxf(gcnt[g], 1.f);
  const float* gs = gsum + (size_t)g * H_;
  float s = 0.f;
#pragma unroll
  for (int k = 0; k < H_; ++k) s = fmaf(gs[k] / c, Wg[k], s);
  out[N + g] = s + bg[0];
}

// ---------------- one GATv2 layer ----------------
static void run_layer(const float* X, int Fin, const float* Wl, const float* Wr,
                      const float* We, const float* att, const float* b,
                      const float* ea, const float* mean, const int* ei,
                      int N, int E, int EP,
                      float* xl, float* xr, float* acc, float* sc,
                      unsigned* nmax, float* nsum, hipStream_t stream) {
  const size_t NH = (size_t)N * H_;
  const int Mt = N >> 4;                 // full 16-row tiles
  if (Fin == FN_) {
    k_gemm2_wmma<FN_><<<Mt, 32, 0, stream>>>(X, Wl, Wr, xl, xr);
  } else {
    k_gemm2_wmma<H_><<<Mt, 32, 0, stream>>>(X, Wl, Wr, xl, xr);
  }
  if (N & 15) {                          // tail rows (none for N=50000)
    k_gemm2_tail<<<N & 15, H_, 0, stream>>>(X, Wl, Wr, xl, xr, N, Fin, Mt << 4);
  }
  k_zero<<<(unsigned)((NH + 255) / 256), 256, 0, stream>>>(acc, NH);
  k_zero<<<(N + 255) / 256, 256, 0, stream>>>((float*)nmax, (size_t)N);
  k_zero<<<(N + 255) / 256, 256, 0, stream>>>(nsum, (size_t)N);
  k_edge_score<<<(EP + 255) / 256, 256, 0, stream>>>(xl, xr, ea, mean, We, att, ei, E, EP, sc, nmax);
  k_edge_exp<<<(EP + 255) / 256, 256, 0, stream>>>(sc, nmax, nsum, ei, E, EP);
  k_edge_agg<<<(EP + 7) / 8, 256, 0, stream>>>(xl, sc, nsum, ei, E, EP, acc);
  k_finalize<<<(unsigned)((NH + 255) / 256), 256, 0, stream>>>(acc, b, NH);
}

extern "C" void kernel_launch(void* const* d_in, const int* in_sizes, int n_in,
                              void* d_out, int out_size, void* d_ws, size_t ws_size,
                              hipStream_t stream) {
  const float* x     = (const float*)d_in[0];
  const int*   ei    = (const int*)d_in[1];
  const float* ea    = (const float*)d_in[2];
  const int*   batch = (const int*)d_in[3];
  const float* Wl1   = (const float*)d_in[4];
  const float* Wr1   = (const float*)d_in[5];
  const float* We1   = (const float*)d_in[6];
  const float* att1  = (const float*)d_in[7];
  const float* b1    = (const float*)d_in[8];
  const float* Wl2   = (const float*)d_in[9];
  const float* Wr2   = (const float*)d_in[10];
  const float* We2   = (const float*)d_in[11];
  const float* att2  = (const float*)d_in[12];
  const float* b2    = (const float*)d_in[13];
  const float* Wn    = (const float*)d_in[14];
  const float* bn    = (const float*)d_in[15];
  const float* Wg    = (const float*)d_in[16];
  const float* bg    = (const float*)d_in[17];
  float* out = (float*)d_out;

  const int N  = in_sizes[0] / FN_;
  const int E  = in_sizes[1] / 2;
  const int EP = E + N;
  const size_t NH = (size_t)N * H_;

  // workspace layout (fp32 slots)
  float* ws   = (float*)d_ws;
  float* xl   = ws;                 // [N*H]
  float* xr   = xl + NH;            // [N*H]
  float* h1   = xr + NH;            // [N*H]  layer-1 accumulator -> h1
  float* h2   = h1 + NH;            // [N*H]  layer-2 accumulator -> h2
  float* sc   = h2 + NH;            // [EP]   per-edge score / exp
  unsigned* nmax = (unsigned*)(sc + EP); // [N] encoded segment max
  float* nsum = (float*)(nmax + N); // [N]
  float* gsum = nsum + N;           // [G*H]
  float* gcnt = gsum + (size_t)G_ * H_; // [G]
  float* mean = gcnt + G_;          // [FE]

  // edge_attr mean (PyG fill_value='mean' for self loops)
  k_zero<<<1, 64, 0, stream>>>(mean, (size_t)FE_);
  k_mean_accum<<<64, 256, 0, stream>>>(ea, E, mean);
  k_mean_scale<<<1, FE_, 0, stream>>>(mean, E);

  // layer 1: x [N,128] -> h1 [N,64]
  run_layer(x, FN_, Wl1, Wr1, We1, att1, b1, ea, mean, ei, N, E, EP,
            xl, xr, h1, sc, nmax, nsum, stream);
  // layer 2: h1 [N,64] -> h2 [N,64]
  run_layer(h1, H_, Wl2, Wr2, We2, att2, b2, ea, mean, ei, N, E, EP,
            xl, xr, h2, sc, nmax, nsum, stream);

  // node head
  k_node_pred<<<(N + 255) / 256, 256, 0, stream>>>(h2, Wn, bn, out, N);

  // graph head (global mean pool)
  k_zero<<<(unsigned)(((size_t)G_ * H_ + G_ + 255) / 256), 256, 0, stream>>>(gsum, (size_t)G_ * H_ + G_);
  k_pool<<<(N + 7) / 8, 256, 0, stream>>>(h2, batch, gsum, gcnt, N);
  k_graph_pred<<<1, G_, 0, stream>>>(gsum, gcnt, Wg, bg, out, N, G_);
}